// Net_44856638439807
// MI455X (gfx1250) — compile-verified
//
#include <hip/hip_runtime.h>
#include <math.h>

typedef float v2f __attribute__((ext_vector_type(2)));
typedef float v8f __attribute__((ext_vector_type(8)));

#define N_NODES_DEF 100000
#define HID 16

// ---------------------------------------------------------------------------
// Workspace layout (floats, n = N_NODES):
//   dis  : [0, n)        degree then rsqrt(degree), in place
//   g1   : [n, 17n)      dis[i] * (x @ W1)   (16 per node)
//   acc1 : [17n, 33n)    scatter accumulator layer 1
//   g2   : [33n, 37n)    dis[i] * (h1_relu @ W2)  (4 per node)
//   acc2 : [37n, 41n)    scatter accumulator layer 2
// ---------------------------------------------------------------------------

__global__ void init_ws_kernel(float* __restrict__ ws, int n) {
    long long i = (long long)blockIdx.x * blockDim.x + threadIdx.x;
    long long nn = n;
    if (i < nn) {
        ws[i] = 1.0f;                        // deg starts at 1 (self loop)
    } else if (i < 17LL * nn) {
        ws[17LL * nn + (i - nn)] = 0.0f;     // acc1
    } else if (i < 21LL * nn) {
        ws[37LL * nn + (i - 17LL * nn)] = 0.0f;  // acc2
    }
}

__global__ void degree_kernel(const int* __restrict__ dst, float* __restrict__ deg, int E) {
    int e = blockIdx.x * blockDim.x + threadIdx.x;
    if (e >= E) return;
    atomicAdd(&deg[dst[e]], 1.0f);
}

__global__ void rsqrt_kernel(float* __restrict__ deg, int n) {
    int i = blockIdx.x * blockDim.x + threadIdx.x;
    if (i >= n) return;
    float d = deg[i];
    deg[i] = (d > 0.0f) ? rsqrtf(d) : 0.0f;
}

// Layer 1 transform: g1[node][:] = dis[node] * (x[node][0:4] @ W1[4][16])
// One wave of 32 lanes handles a 16-node tile with a single f32 WMMA.
__global__ void xw1_wmma_kernel(const float* __restrict__ x,
                                const float* __restrict__ W1,
                                const float* __restrict__ dis,
                                float* __restrict__ g1, int n_tiles) {
    int wave = (blockIdx.x * blockDim.x + threadIdx.x) >> 5;
    int lane = threadIdx.x & 31;
    if (wave >= n_tiles) return;
    int m  = lane & 15;          // row within tile (A) / column N (B, C/D)
    int hb = lane >> 4;          // half-wave: K offset selector
    int kb = 2 * hb;             // K base for this half (0 or 2)
    int node = wave * 16 + m;

    // A fragment: A[m][kb+v] = x[node][kb+v]
    v2f a;
    a.x = x[(size_t)node * 4 + kb + 0];
    a.y = x[(size_t)node * 4 + kb + 1];
    // B fragment: B[kb+v][m] = W1[(kb+v)*16 + m]
    v2f b;
    b.x = W1[(kb + 0) * 16 + m];
    b.y = W1[(kb + 1) * 16 + m];

    v8f c = {};
    c = __builtin_amdgcn_wmma_f32_16x16x4_f32(false, a, false, b, (short)0, c,
                                              false, false);

    // D: VGPR r holds row (r + 8*hb), col m
#pragma unroll
    for (int r = 0; r < 8; ++r) {
        int row = wave * 16 + r + 8 * hb;
        g1[(size_t)row * HID + m] = dis[row] * c[r];
    }
}

// Edge scatter layer 1: acc1[dst] += g1[src]  (16 floats / edge)
__global__ void agg1_kernel(const int* __restrict__ src, const int* __restrict__ dst,
                            const float* __restrict__ g1, float* __restrict__ acc1,
                            int E) {
    int e = blockIdx.x * blockDim.x + threadIdx.x;
    if (e >= E) return;
    int s = src[e];
    int d = dst[e];
    const float4* gs = (const float4*)(g1 + (size_t)s * HID);
    float* ad = acc1 + (size_t)d * HID;
#pragma unroll
    for (int q = 0; q < 4; ++q) {
        float4 v = gs[q];
        atomicAdd(ad + 4 * q + 0, v.x);
        atomicAdd(ad + 4 * q + 1, v.y);
        atomicAdd(ad + 4 * q + 2, v.z);
        atomicAdd(ad + 4 * q + 3, v.w);
    }
}

// Layer-1 epilogue fused with layer-2 transform:
//   h1r[i][k] = relu(dis[i]*(acc1[i][k] + g1[i][k]) + b1[k])   (built in-register
//   directly in WMMA A-fragment layout), then
//   g2[i][0:4] = dis[i] * (h1r[i][:] @ W2[16][4])   via 4 chained K=4 WMMAs
//   (W2 zero-padded from 4 to 16 output columns).
__global__ void layer2_wmma_kernel(const float* __restrict__ acc1,
                                   const float* __restrict__ g1,
                                   const float* __restrict__ dis,
                                   const float* __restrict__ b1,
                                   const float* __restrict__ W2,
                                   float* __restrict__ g2, int n_tiles) {
    int wave = (blockIdx.x * blockDim.x + threadIdx.x) >> 5;
    int lane = threadIdx.x & 31;
    if (wave >= n_tiles) return;
    int m  = lane & 15;
    int hb = lane >> 4;
    int node = wave * 16 + m;
    float ds = dis[node];

    v8f c = {};
#pragma unroll
    for (int s4 = 0; s4 < 4; ++s4) {
        int kb = 4 * s4 + 2 * hb;  // global K index of first of the two A elems
        // A fragment: relu(dis*(acc1+g1)+b1) for features kb, kb+1 of `node`
        float t0 = fmaf(ds, acc1[(size_t)node * HID + kb + 0] + g1[(size_t)node * HID + kb + 0], b1[kb + 0]);
        float t1 = fmaf(ds, acc1[(size_t)node * HID + kb + 1] + g1[(size_t)node * HID + kb + 1], b1[kb + 1]);
        v2f a;
        a.x = t0 > 0.0f ? t0 : 0.0f;
        a.y = t1 > 0.0f ? t1 : 0.0f;
        // B fragment: W2 (16x4) zero-padded to 16 columns; B[k][n] = W2[k*4+n]
        v2f b;
        b.x = (m < 4) ? W2[(kb + 0) * 4 + m] : 0.0f;
        b.y = (m < 4) ? W2[(kb + 1) * 4 + m] : 0.0f;
        c = __builtin_amdgcn_wmma_f32_16x16x4_f32(false, a, false, b, (short)0, c,
                                                  false, false);
    }

    if (m < 4) {
#pragma unroll
        for (int r = 0; r < 8; ++r) {
            int row = wave * 16 + r + 8 * hb;
            g2[(size_t)row * 4 + m] = dis[row] * c[r];
        }
    }
}

// Edge scatter layer 2: acc2[dst] += g2[src]  (4 floats / edge)
__global__ void agg2_kernel(const int* __restrict__ src, const int* __restrict__ dst,
                            const float* __restrict__ g2, float* __restrict__ acc2,
                            int E) {
    int e = blockIdx.x * blockDim.x + threadIdx.x;
    if (e >= E) return;
    int s = src[e];
    int d = dst[e];
    float4 v = *(const float4*)(g2 + (size_t)s * 4);
    float* ad = acc2 + (size_t)d * 4;
    atomicAdd(ad + 0, v.x);
    atomicAdd(ad + 1, v.y);
    atomicAdd(ad + 2, v.z);
    atomicAdd(ad + 3, v.w);
}

// Final: out = log_softmax(dis*(acc2+g2) + b2) over 4 classes
__global__ void finalize_kernel(const float* __restrict__ acc2,
                                const float* __restrict__ g2,
                                const float* __restrict__ dis,
                                const float* __restrict__ b2,
                                float* __restrict__ out, int n) {
    int i = blockIdx.x * blockDim.x + threadIdx.x;
    if (i >= n) return;
    float ds = dis[i];
    float v[4];
#pragma unroll
    for (int j = 0; j < 4; ++j)
        v[j] = fmaf(ds, acc2[(size_t)i * 4 + j] + g2[(size_t)i * 4 + j], b2[j]);
    float mx = fmaxf(fmaxf(v[0], v[1]), fmaxf(v[2], v[3]));
    float sum = expf(v[0] - mx) + expf(v[1] - mx) + expf(v[2] - mx) + expf(v[3] - mx);
    float lse = logf(sum);
#pragma unroll
    for (int j = 0; j < 4; ++j)
        out[(size_t)i * 4 + j] = v[j] - mx - lse;
}

extern "C" void kernel_launch(void* const* d_in, const int* in_sizes, int n_in,
                              void* d_out, int out_size, void* d_ws, size_t ws_size,
                              hipStream_t stream) {
    (void)n_in; (void)out_size; (void)ws_size;
    const float* x   = (const float*)d_in[0];   // [N,4]
    const float* W1  = (const float*)d_in[1];   // [4,16]
    const float* b1  = (const float*)d_in[2];   // [16]
    const float* W2  = (const float*)d_in[3];   // [16,4]
    const float* b2  = (const float*)d_in[4];   // [4]
    const int*   ei  = (const int*)d_in[5];     // [2,E]
    float* out = (float*)d_out;

    const int n = in_sizes[0] / 4;              // N_NODES
    const int E = in_sizes[5] / 2;              // N_EDGES
    const int* src = ei;
    const int* dst = ei + E;

    float* ws   = (float*)d_ws;
    const size_t nn = (size_t)n;
    float* dis  = ws;
    float* g1   = ws + nn;
    float* acc1 = ws + 17 * nn;
    float* g2   = ws + 33 * nn;
    float* acc2 = ws + 37 * nn;

    const int B = 256;
    const long long init_total = 21LL * n;
    const int n_tiles = (n + 15) / 16;
    const int wmma_threads = n_tiles * 32;

    init_ws_kernel<<<(int)((init_total + B - 1) / B), B, 0, stream>>>(ws, n);
    degree_kernel<<<(E + B - 1) / B, B, 0, stream>>>(dst, dis, E);
    rsqrt_kernel<<<(n + B - 1) / B, B, 0, stream>>>(dis, n);
    xw1_wmma_kernel<<<(wmma_threads + B - 1) / B, B, 0, stream>>>(x, W1, dis, g1, n_tiles);
    agg1_kernel<<<(E + B - 1) / B, B, 0, stream>>>(src, dst, g1, acc1, E);
    layer2_wmma_kernel<<<(wmma_threads + B - 1) / B, B, 0, stream>>>(acc1, g1, dis, b1, W2, g2, n_tiles);
    agg2_kernel<<<(E + B - 1) / B, B, 0, stream>>>(src, dst, g2, acc2, E);
    finalize_kernel<<<(n + B - 1) / B, B, 0, stream>>>(acc2, g2, dis, b2, out, n);
}